// RAAC_89541478187505
// MI455X (gfx1250) — compile-verified
//
#include <hip/hip_runtime.h>
#include <hip/hip_bf16.h>
#include <cstddef>
#include <cstdint>

// Problem geometry (feature: [32, 512, 64, 64] fp32)
#define HW    4096                 // 64*64 contiguous floats per (b,c) row
#define CCH   512
#define NBAT  32
#define ROWS  (NBAT * CCH)         // 16384 rows
#define WPB   8                    // waves per block (wave32)
#define NBUF  8                    // LDS ring buffers per wave
#define CHUNK 128                  // floats staged per wave per step (32 lanes * float4)
#define ITERS (HW / CHUNK)         // 32 pipeline steps per row
#define DEPTH 4                    // async stages in flight

// CDNA5 async DMA: per-lane 16B global -> LDS, tracked by ASYNCcnt.
__device__ __forceinline__ void async_ld_b128(unsigned lds_off, const float* gaddr) {
    asm volatile("global_load_async_to_lds_b128 %0, %1, off"
                 :: "v"(lds_off), "v"(gaddr) : "memory");
}

// rotl1: unsigned compare of rotl(bits,1) orders by |x| first, sign last.
__device__ __forceinline__ unsigned key_of(float f) {
    unsigned b = __float_as_uint(f);
    return (b << 1) | (b >> 31);
}
__device__ __forceinline__ unsigned umax(unsigned a, unsigned b) { return a > b ? a : b; }

__global__ __launch_bounds__(WPB * 32)
void rowmax_kernel(const float* __restrict__ feat, float* __restrict__ xout) {
    // 8 waves * 8 buffers * 512B = 32 KiB staged LDS per block (of 320 KiB/WGP)
    __shared__ __align__(16) float stage[WPB * NBUF * CHUNK];

    const int lane = threadIdx.x & 31;
    const int wave = threadIdx.x >> 5;
    const int row  = blockIdx.x * WPB + wave;          // row = b*512 + c
    const float* rowp = feat + (size_t)row * HW;

    float* mybuf = &stage[wave * NBUF * CHUNK];
    // Generic pointer to LDS: low 32 bits are the workgroup-relative LDS byte offset.
    const unsigned lds_base = (unsigned)(size_t)(void*)mybuf;

    // Prologue: fill the pipeline (stages 0..DEPTH-1)
#pragma unroll
    for (int j = 0; j < DEPTH; ++j) {
        async_ld_b128(lds_base + (unsigned)((j & (NBUF - 1)) * (CHUNK * 4) + lane * 16),
                      rowp + j * CHUNK + lane * 4);
    }

    unsigned bestkey = 0u;
#pragma unroll
    for (int i = 0; i < ITERS; ++i) {
        // Issue stage i+DEPTH into a ring slot last read at step i-DEPTH (long done).
        if (i + DEPTH < ITERS) {
            const int j = i + DEPTH;
            async_ld_b128(lds_base + (unsigned)((j & (NBUF - 1)) * (CHUNK * 4) + lane * 16),
                          rowp + j * CHUNK + lane * 4);
        }
        // Async loads complete in order: issued = min(i+5,32); stage i done once
        // ASYNCcnt <= issued-(i+1)  ==>  4,4,...,4,3,2,1,0 across the tail.
        if (i < ITERS - DEPTH)    asm volatile("s_wait_asynccnt 4" ::: "memory");
        else if (i == ITERS - 4)  asm volatile("s_wait_asynccnt 3" ::: "memory");
        else if (i == ITERS - 3)  asm volatile("s_wait_asynccnt 2" ::: "memory");
        else if (i == ITERS - 2)  asm volatile("s_wait_asynccnt 1" ::: "memory");
        else                      asm volatile("s_wait_asynccnt 0" ::: "memory");

        const float4 v = ((const float4*)(mybuf + (i & (NBUF - 1)) * CHUNK))[lane];
        bestkey = umax(bestkey, key_of(v.x));
        bestkey = umax(bestkey, key_of(v.y));
        bestkey = umax(bestkey, key_of(v.z));
        bestkey = umax(bestkey, key_of(v.w));
    }

    // wave32 cross-lane max (no LDS, no barriers)
#pragma unroll
    for (int off = 16; off > 0; off >>= 1) {
        unsigned o = (unsigned)__shfl_xor((int)bestkey, off, 32);
        bestkey = umax(bestkey, o);
    }

    if (lane == 0) {
        unsigned bits = (bestkey >> 1) | (bestkey << 31);   // rotr1: recover signed value
        xout[row] = __uint_as_float(bits);
    }
}

__global__ __launch_bounds__(CCH)
void l2norm_kernel(const float* __restrict__ x, float* __restrict__ out) {
    const int b    = blockIdx.x;
    const int c    = threadIdx.x;
    const int lane = c & 31;
    const int wave = c >> 5;

    const float v = x[b * CCH + c];
    float s = v * v;
#pragma unroll
    for (int off = 16; off > 0; off >>= 1) s += __shfl_xor(s, off, 32);

    __shared__ float part[CCH / 32];
    __shared__ float norm_s;
    if (lane == 0) part[wave] = s;
    __syncthreads();
    if (wave == 0) {
        float t = (lane < CCH / 32) ? part[lane] : 0.0f;
#pragma unroll
        for (int off = 8; off > 0; off >>= 1) t += __shfl_xor(t, off, 32);
        if (lane == 0) norm_s = sqrtf(t);
    }
    __syncthreads();

    out[b * CCH + c] = v / fmaxf(norm_s, 1e-12f);   // F.normalize eps clamp
}

extern "C" void kernel_launch(void* const* d_in, const int* in_sizes, int n_in,
                              void* d_out, int out_size, void* d_ws, size_t ws_size,
                              hipStream_t stream) {
    (void)in_sizes; (void)n_in; (void)out_size; (void)ws_size;
    const float* feat = (const float*)d_in[0];
    float* xbuf = (float*)d_ws;      // 16384 floats of scratch (64 KiB)
    float* out  = (float*)d_out;     // [32, 512] fp32

    rowmax_kernel<<<ROWS / WPB, WPB * 32, 0, stream>>>(feat, xbuf);
    l2norm_kernel<<<NBAT, CCH, 0, stream>>>(xbuf, out);
}